// AttentionPooling_52037823758337
// MI455X (gfx1250) — compile-verified
//
#include <hip/hip_runtime.h>
#include <hip/hip_bf16.h>

#define NTOK 524288
#define BSEG 1024
#define DIM  256
#define NH   4
#define DH   64

typedef float v2f __attribute__((ext_vector_type(2)));
typedef float v8f __attribute__((ext_vector_type(8)));

// ---------------------------------------------------------------------------
// Kernel 1: one block. q = LN(seed); qh = (q@w_q^T + b_q)*scale;
// a[h][c] = sum_j w_k[h*64+j][c] * qh[h][j];  cb[h] = sum_j b_k[h*64+j]*qh[h][j]
// ---------------------------------------------------------------------------
__global__ __launch_bounds__(256) void prep_kernel(
    const float* __restrict__ seed, const float* __restrict__ ln_q_w,
    const float* __restrict__ ln_q_b, const float* __restrict__ w_q,
    const float* __restrict__ b_q, const float* __restrict__ w_k,
    const float* __restrict__ b_k, float* __restrict__ a_out,
    float* __restrict__ cb_out) {
  __shared__ float sq[DIM];
  __shared__ float sqh[DIM];
  __shared__ float red[16];
  int t = threadIdx.x;
  float v = seed[t];
  float s = v, ss = v * v;
#pragma unroll
  for (int off = 16; off; off >>= 1) {
    s += __shfl_xor(s, off, 32);
    ss += __shfl_xor(ss, off, 32);
  }
  int wave = t >> 5, lane = t & 31;
  if (lane == 0) { red[wave] = s; red[8 + wave] = ss; }
  __syncthreads();
  float tot = 0.f, tot2 = 0.f;
#pragma unroll
  for (int w = 0; w < 8; ++w) { tot += red[w]; tot2 += red[8 + w]; }
  float mean = tot * (1.f / DIM);
  float var  = tot2 * (1.f / DIM) - mean * mean;
  float rinv = rsqrtf(var + 1e-5f);
  sq[t] = (v - mean) * rinv * ln_q_w[t] + ln_q_b[t];
  __syncthreads();
  float acc = b_q[t];
  for (int c = 0; c < DIM; ++c) acc = fmaf(sq[c], w_q[t * DIM + c], acc);
  sqh[t] = acc * 0.125f;  // 1/sqrt(DH=64)
  __syncthreads();
#pragma unroll
  for (int h = 0; h < NH; ++h) {
    float aa = 0.f;
    for (int j = 0; j < DH; ++j)
      aa = fmaf(w_k[(h * DH + j) * DIM + t], sqh[h * DH + j], aa);
    a_out[h * DIM + t] = aa;
  }
  if (t < NH) {
    float c2 = 0.f;
    for (int j = 0; j < DH; ++j) c2 = fmaf(b_k[t * DH + j], sqh[t * DH + j], c2);
    cb_out[t] = c2;
  }
}

// ---------------------------------------------------------------------------
// Kernel 2: segment starts from sorted batch ids. segstart[b]..segstart[b+1]
// ---------------------------------------------------------------------------
__global__ void segstart_kernel(const int* __restrict__ batch,
                                int* __restrict__ segstart) {
  int n = blockIdx.x * blockDim.x + threadIdx.x;
  if (n >= NTOK) return;
  int cur = batch[n];
  int prev = (n == 0) ? -1 : batch[n - 1];
  for (int b = prev + 1; b <= cur; ++b) segstart[b] = n;
  if (n == NTOK - 1)
    for (int b = cur + 1; b <= BSEG; ++b) segstart[b] = NTOK;
}

// ---------------------------------------------------------------------------
// Kernel 3: one block per segment. Streaming pass over x: LN, 4 scores via
// precomputed a/cb, per-wave ONLINE softmax (flash-style), merge 8 waves.
// Writes Sg[b][h][c] = (sum e*kv)/den  and nonempty[b].
// ---------------------------------------------------------------------------
__global__ __launch_bounds__(256) void pool_kernel(
    const float* __restrict__ x, const int* __restrict__ segstart,
    const float* __restrict__ ln_k_w, const float* __restrict__ ln_k_b,
    const float* __restrict__ a_vec, const float* __restrict__ cb,
    float* __restrict__ Sg, float* __restrict__ nonempty) {
  __shared__ float ldsS[8][NH][DIM];   // 32 KB
  __shared__ float ldsM[8][NH];
  __shared__ float ldsD[8][NH];
  __shared__ float wfac[8][NH];
  __shared__ float invden_s[NH];
  int b = blockIdx.x;
  int tid = threadIdx.x;
  int lane = tid & 31, wave = tid >> 5;
  int s0 = segstart[b], s1 = segstart[b + 1];
  int len = s1 - s0;
  if (len == 0) {  // reference: den guarded to 1 => pooled contribution is 0
    for (int k = tid; k < NH * DIM; k += 256) Sg[(size_t)b * NH * DIM + k] = 0.f;
    if (tid == 0) nonempty[b] = 0.f;
    return;
  }
  int cbase = lane * 8;
  float g[8], bb[8], areg[NH][8], cbl[NH];
#pragma unroll
  for (int i = 0; i < 8; ++i) { g[i] = ln_k_w[cbase + i]; bb[i] = ln_k_b[cbase + i]; }
#pragma unroll
  for (int h = 0; h < NH; ++h) {
#pragma unroll
    for (int i = 0; i < 8; ++i) areg[h][i] = a_vec[h * DIM + cbase + i];
    cbl[h] = cb[h];
  }
  float m[NH], den[NH], S[NH][8];
#pragma unroll
  for (int h = 0; h < NH; ++h) {
    m[h] = -__builtin_inff(); den[h] = 0.f;
#pragma unroll
    for (int i = 0; i < 8; ++i) S[h][i] = 0.f;
  }
  for (int t = wave; t < len; t += 8) {
    const float* xp = x + (size_t)(s0 + t) * DIM + cbase;
    __builtin_prefetch(xp + 8 * DIM, 0, 0);  // global_prefetch_b8, next token
    float4 x0 = ((const float4*)xp)[0];
    float4 x1 = ((const float4*)xp)[1];
    float xv[8] = {x0.x, x0.y, x0.z, x0.w, x1.x, x1.y, x1.z, x1.w};
    float s = 0.f, ss = 0.f;
#pragma unroll
    for (int i = 0; i < 8; ++i) { s += xv[i]; ss = fmaf(xv[i], xv[i], ss); }
#pragma unroll
    for (int off = 16; off; off >>= 1) {
      s += __shfl_xor(s, off, 32);
      ss += __shfl_xor(ss, off, 32);
    }
    float mean = s * (1.f / DIM);
    float var  = ss * (1.f / DIM) - mean * mean;
    float rinv = rsqrtf(var + 1e-5f);
    float kv[8];
#pragma unroll
    for (int i = 0; i < 8; ++i) kv[i] = (xv[i] - mean) * rinv * g[i] + bb[i];
    float p[NH];
#pragma unroll
    for (int h = 0; h < NH; ++h) {
      float acc = 0.f;
#pragma unroll
      for (int i = 0; i < 8; ++i) acc = fmaf(kv[i], areg[h][i], acc);
      p[h] = acc;
    }
#pragma unroll
    for (int h = 0; h < NH; ++h)
#pragma unroll
      for (int off = 16; off; off >>= 1) p[h] += __shfl_xor(p[h], off, 32);
#pragma unroll
    for (int h = 0; h < NH; ++h) {
      float sc = p[h] + cbl[h];      // wave-uniform
      if (sc <= m[h]) {              // common path: no rescale
        float e = __expf(sc - m[h]);
        den[h] += e;
#pragma unroll
        for (int i = 0; i < 8; ++i) S[h][i] = fmaf(e, kv[i], S[h][i]);
      } else {                       // rare: new running max
        float r = __expf(m[h] - sc); // exp(-inf)=0 handles first token
        den[h] = fmaf(den[h], r, 1.f);
#pragma unroll
        for (int i = 0; i < 8; ++i) S[h][i] = fmaf(S[h][i], r, kv[i]);
        m[h] = sc;
      }
    }
  }
#pragma unroll
  for (int h = 0; h < NH; ++h) {
#pragma unroll
    for (int i = 0; i < 8; ++i) ldsS[wave][h][cbase + i] = S[h][i];
    if (lane == 0) { ldsM[wave][h] = m[h]; ldsD[wave][h] = den[h]; }
  }
  __syncthreads();
  if (tid < 32) {
    int w = tid >> 2, h = tid & 3;
    float M = -__builtin_inff();
    for (int ww = 0; ww < 8; ++ww) M = fmaxf(M, ldsM[ww][h]);
    wfac[w][h] = __expf(ldsM[w][h] - M);  // waves with no tokens -> 0
  }
  __syncthreads();
  if (tid < NH) {
    float dt = 0.f;
    for (int ww = 0; ww < 8; ++ww) dt = fmaf(ldsD[ww][tid], wfac[ww][tid], dt);
    invden_s[tid] = (dt > 0.f) ? (1.f / dt) : 0.f;
  }
  __syncthreads();
  int c = tid;
#pragma unroll
  for (int h = 0; h < NH; ++h) {
    float tot = 0.f;
    for (int ww = 0; ww < 8; ++ww) tot = fmaf(ldsS[ww][h][c], wfac[ww][h], tot);
    Sg[(size_t)b * NH * DIM + h * DIM + c] = tot * invden_s[h];
  }
  if (tid == 0) nonempty[b] = 1.f;
}

// ---------------------------------------------------------------------------
// Kernel 4 (WMMA f32 16x16x4): pooled[b][h*64+j] = Sg[b][h][:]·w_v[h*64+j][:]
//                                                 + b_v[h*64+j]*nonempty[b]
// One wave per 16x16 tile; K=256 -> 64 WMMA. A/B fragments are float2 loads
// matching the ISA layout (lane: M=lane%16, K = vgpr + 2*(lane>=16)).
// ---------------------------------------------------------------------------
__global__ __launch_bounds__(256) void gemm_v_kernel(
    const float* __restrict__ Sg, const float* __restrict__ w_v,
    const float* __restrict__ b_v, const float* __restrict__ nonempty,
    float* __restrict__ pooled) {
  int wid = (blockIdx.x * blockDim.x + threadIdx.x) >> 5;  // 0..1023
  int lane = threadIdx.x & 31;
  int h = wid >> 8;
  int rem = wid & 255;
  int b0 = (rem >> 2) * 16;
  int j0 = (rem & 3) * 16;
  int halfSel = lane >> 4;
  int lm = lane & 15;
  const float* Arow = Sg + (size_t)(b0 + lm) * (NH * DIM) + h * DIM;
  const float* Brow = w_v + (size_t)(h * DH + j0 + lm) * DIM;
  v8f acc = {};
  for (int kk = 0; kk < DIM; kk += 4) {
    int kb = kk + 2 * halfSel;
    v2f av = *(const v2f*)(Arow + kb);
    v2f bv = *(const v2f*)(Brow + kb);
    acc = __builtin_amdgcn_wmma_f32_16x16x4_f32(false, av, false, bv,
                                                (short)0, acc, false, false);
  }
  float bcol = b_v[h * DH + j0 + lm];
#pragma unroll
  for (int r = 0; r < 8; ++r) {
    int mrow = b0 + r + 8 * halfSel;
    pooled[(size_t)mrow * DIM + h * DH + j0 + lm] =
        acc[r] + bcol * nonempty[mrow];
  }
}

// ---------------------------------------------------------------------------
// Kernel 5 (WMMA f32 16x16x4): out = pooled @ w_o^T + b_o   (1024x256x256)
// ---------------------------------------------------------------------------
__global__ __launch_bounds__(256) void gemm_o_kernel(
    const float* __restrict__ pooled, const float* __restrict__ w_o,
    const float* __restrict__ b_o, float* __restrict__ out) {
  int wid = (blockIdx.x * blockDim.x + threadIdx.x) >> 5;  // 0..1023
  int lane = threadIdx.x & 31;
  int b0 = (wid >> 4) * 16;
  int n0 = (wid & 15) * 16;
  int halfSel = lane >> 4;
  int lm = lane & 15;
  const float* Arow = pooled + (size_t)(b0 + lm) * DIM;
  const float* Brow = w_o + (size_t)(n0 + lm) * DIM;
  v8f acc = {};
  for (int kk = 0; kk < DIM; kk += 4) {
    int kb = kk + 2 * halfSel;
    v2f av = *(const v2f*)(Arow + kb);
    v2f bv = *(const v2f*)(Brow + kb);
    acc = __builtin_amdgcn_wmma_f32_16x16x4_f32(false, av, false, bv,
                                                (short)0, acc, false, false);
  }
  float bc = b_o[n0 + lm];
#pragma unroll
  for (int r = 0; r < 8; ++r) {
    int mrow = b0 + r + 8 * halfSel;
    out[(size_t)mrow * DIM + n0 + lm] = acc[r] + bc;
  }
}

// ---------------------------------------------------------------------------
extern "C" void kernel_launch(void* const* d_in, const int* in_sizes, int n_in,
                              void* d_out, int out_size, void* d_ws,
                              size_t ws_size, hipStream_t stream) {
  const float* x      = (const float*)d_in[0];
  const int*   batch  = (const int*)d_in[1];
  const float* seed   = (const float*)d_in[2];
  const float* ln_q_w = (const float*)d_in[3];
  const float* ln_q_b = (const float*)d_in[4];
  const float* ln_k_w = (const float*)d_in[5];
  const float* ln_k_b = (const float*)d_in[6];
  const float* w_q    = (const float*)d_in[7];
  const float* b_q    = (const float*)d_in[8];
  const float* w_k    = (const float*)d_in[9];
  const float* b_k    = (const float*)d_in[10];
  const float* w_v    = (const float*)d_in[11];
  const float* b_v    = (const float*)d_in[12];
  const float* w_o    = (const float*)d_in[13];
  const float* b_o    = (const float*)d_in[14];
  float* out = (float*)d_out;

  char* ws = (char*)d_ws;
  float* a_vec    = (float*)(ws + 0);        // 4 KB
  float* cb       = (float*)(ws + 4096);     // 16 B
  float* nonempty = (float*)(ws + 4352);     // 4 KB
  int*   segstart = (int*)(ws + 8448);       // 4.1 KB
  float* Sg       = (float*)(ws + 16384);    // 4 MB
  float* pooled   = (float*)(ws + 16384 + (size_t)BSEG * NH * DIM * 4);  // 1 MB

  prep_kernel<<<1, 256, 0, stream>>>(seed, ln_q_w, ln_q_b, w_q, b_q, w_k, b_k,
                                     a_vec, cb);
  segstart_kernel<<<NTOK / 256, 256, 0, stream>>>(batch, segstart);
  pool_kernel<<<BSEG, 256, 0, stream>>>(x, segstart, ln_k_w, ln_k_b, a_vec, cb,
                                        Sg, nonempty);
  gemm_v_kernel<<<128, 256, 0, stream>>>(Sg, w_v, b_v, nonempty, pooled);
  gemm_o_kernel<<<128, 256, 0, stream>>>(pooled, w_o, b_o, out);
}